// DLUPack_53008486367289
// MI455X (gfx1250) — compile-verified
//
#include <hip/hip_runtime.h>
#include <hip/hip_bf16.h>

// ---------------------------------------------------------------------------
// DLUPack (CARAFE-style deformable upsampler), fp32, gfx1250 / wave32.
// N=2, C=256, H=W=64, S=2, K=5, CC=64, EK=3, HOUT=WOUT=128
// ---------------------------------------------------------------------------

#define NB   2
#define CIN  256
#define HH   64
#define WW   64
#define PIX  (HH * WW)          // 4096
#define CC   64
#define NOC  33                 // 8 offset + 25 kernel channels
#define KK   25
#define HOUT 128
#define WOUT 128

typedef __attribute__((ext_vector_type(2))) float v2f;
typedef __attribute__((ext_vector_type(8))) float v8f;

// ---------------------------------------------------------------------------
// Stage 1: comp[n][cc][p] = b[cc] + sum_c w[cc][c] * x[n][c][p]
// GEMM M=64, K=256, N=4096 per batch via V_WMMA_F32_16X16X4_F32.
// One wave: all 4 M-tiles for one 16-pixel N-tile (B fragment reused 4x).
// A layout (16x4 f32): lane<16 -> M=lane, v0=K0,v1=K1 ; lane>=16 -> K2,K3.
// B layout (4x16 f32): lane%16 = N, low half rows K0/K1, high half K2/K3.
// C/D layout: VGPR r -> M = r + 8*(lane>=16), N = lane%16.
// ---------------------------------------------------------------------------
__global__ __launch_bounds__(128)
void comp_wmma_kernel(const float* __restrict__ x,
                      const float* __restrict__ w_comp,
                      const float* __restrict__ b_comp,
                      float* __restrict__ comp)
{
    const int lane  = threadIdx.x & 31;
    const int wv    = threadIdx.x >> 5;
    const int gwave = blockIdx.x * 4 + wv;      // 512 waves total
    const int n     = gwave >> 8;               // batch
    const int ntile = gwave & 255;              // 256 N-tiles of 16 pixels
    const int p0    = ntile << 4;

    const int lm = lane & 15;                   // N (for B/C/D) or M (for A)
    const int hi = lane >> 4;                   // K-half selector

    const float* xb = x + (size_t)n * CIN * PIX;

    v8f acc[4] = {};

#pragma unroll 4
    for (int k = 0; k < CIN; k += 4) {
        const int c0 = k + 2 * hi;
        v2f b;
        b.x = xb[(size_t)c0 * PIX + p0 + lm];
        b.y = xb[(size_t)(c0 + 1) * PIX + p0 + lm];
#pragma unroll
        for (int t = 0; t < 4; ++t) {
            const int m = (t << 4) + lm;
            v2f a;
            a.x = w_comp[m * CIN + c0];
            a.y = w_comp[m * CIN + c0 + 1];
            acc[t] = __builtin_amdgcn_wmma_f32_16x16x4_f32(
                false, a, false, b, (short)0, acc[t], false, false);
        }
    }

#pragma unroll
    for (int t = 0; t < 4; ++t) {
#pragma unroll
        for (int r = 0; r < 8; ++r) {
            const int cc = (t << 4) + r + (hi << 3);
            comp[((size_t)n * CC + cc) * PIX + p0 + lm] = acc[t][r] + b_comp[cc];
        }
    }
}

// ---------------------------------------------------------------------------
// Stage 2: off (8 ch) and ker (25 ch) = 3x3 conv over comp, pad 1.
// ---------------------------------------------------------------------------
__global__ __launch_bounds__(256)
void offker_kernel(const float* __restrict__ comp,
                   const float* __restrict__ w_off, const float* __restrict__ b_off,
                   const float* __restrict__ w_ker, const float* __restrict__ b_ker,
                   float* __restrict__ offb, float* __restrict__ kerb)
{
    const int idx = blockIdx.x * blockDim.x + threadIdx.x;
    if (idx >= NB * NOC * PIX) return;
    const int p  = idx & (PIX - 1);
    const int t  = idx / PIX;
    const int oc = t % NOC;
    const int n  = t / NOC;
    const int h  = p >> 6;
    const int w  = p & 63;

    const float* wgt;
    float s;
    if (oc < 8) { wgt = w_off + oc * CC * 9;        s = b_off[oc]; }
    else        { wgt = w_ker + (oc - 8) * CC * 9;  s = b_ker[oc - 8]; }

    const float* cb = comp + (size_t)n * CC * PIX;

    for (int c = 0; c < CC; ++c) {
        const float* cp = cb + (size_t)c * PIX;
        const float* wp = wgt + c * 9;
        float wr[9];
#pragma unroll
        for (int i = 0; i < 9; ++i) wr[i] = wp[i];
#pragma unroll
        for (int dy = -1; dy <= 1; ++dy) {
            const int y = h + dy;
            if (y < 0 || y > HH - 1) continue;
#pragma unroll
            for (int dx = -1; dx <= 1; ++dx) {
                const int xw = w + dx;
                if (xw < 0 || xw > WW - 1) continue;
                s += cp[y * WW + xw] * wr[(dy + 1) * 3 + (dx + 1)];
            }
        }
    }
    if (oc < 8) offb[((size_t)n * 8 + oc) * PIX + p]        = s;
    else        kerb[((size_t)n * KK + (oc - 8)) * PIX + p] = s;
}

// ---------------------------------------------------------------------------
// Stage 3: softmax over the 25 ker channels (in place).
// ---------------------------------------------------------------------------
__global__ __launch_bounds__(256)
void softmax_kernel(float* __restrict__ kerb)
{
    const int idx = blockIdx.x * blockDim.x + threadIdx.x;
    if (idx >= NB * PIX) return;
    const int n = idx >> 12;
    const int p = idx & (PIX - 1);
    float* base = kerb + (size_t)n * KK * PIX + p;

    float v[KK];
    float mx = -3.4e38f;
#pragma unroll
    for (int k = 0; k < KK; ++k) { v[k] = base[(size_t)k * PIX]; mx = fmaxf(mx, v[k]); }
    float s = 0.f;
#pragma unroll
    for (int k = 0; k < KK; ++k) { v[k] = __expf(v[k] - mx); s += v[k]; }
    const float inv = 1.0f / s;
#pragma unroll
    for (int k = 0; k < KK; ++k) base[(size_t)k * PIX] = v[k] * inv;
}

// ---------------------------------------------------------------------------
// Stage 4: bilinear grid-sample (border, align_corners) of the 25 mask maps.
// Normalized-coord algebra collapses to ix = clamp(w + off_x, 0, W-1).
// ---------------------------------------------------------------------------
__global__ __launch_bounds__(256)
void maskup_kernel(const float* __restrict__ offb,
                   const float* __restrict__ m,      // softmaxed kerb
                   float* __restrict__ maskup)
{
    const int idx = blockIdx.x * blockDim.x + threadIdx.x;
    if (idx >= NB * HOUT * WOUT) return;
    const int xo = idx & 127;
    const int yo = (idx >> 7) & 127;
    const int n  = idx >> 14;
    const int h = yo >> 1, w = xo >> 1;
    const int pq = (yo & 1) * 2 + (xo & 1);

    const float offx = offb[((size_t)n * 8 + pq) * PIX + h * WW + w];
    const float offy = offb[((size_t)n * 8 + 4 + pq) * PIX + h * WW + w];

    const float ix = fminf(fmaxf((float)w + offx, 0.f), (float)(WW - 1));
    const float iy = fminf(fmaxf((float)h + offy, 0.f), (float)(HH - 1));

    const float fx0 = floorf(ix), fy0 = floorf(iy);
    const float fx = ix - fx0,    fy = iy - fy0;
    const int x0i = (int)fx0,     y0i = (int)fy0;
    const int x1i = min(x0i + 1, WW - 1);
    const int y1i = min(y0i + 1, HH - 1);

    const float w00 = (1.f - fx) * (1.f - fy);
    const float w01 = fx * (1.f - fy);
    const float w10 = (1.f - fx) * fy;
    const float w11 = fx * fy;

    const int i00 = y0i * WW + x0i, i01 = y0i * WW + x1i;
    const int i10 = y1i * WW + x0i, i11 = y1i * WW + x1i;

    const float* mb = m + (size_t)n * KK * PIX;
#pragma unroll
    for (int k = 0; k < KK; ++k) {
        const float* mk = mb + (size_t)k * PIX;
        const float v = mk[i00] * w00 + mk[i01] * w01 + mk[i10] * w10 + mk[i11] * w11;
        maskup[(((size_t)n * KK + k) << 14) + (yo << 7) + xo] = v;
    }
}

// ---------------------------------------------------------------------------
// Stage 5: out[n,c,y,x] = sum_{ki,kj} x_pad[n,c,h+ki-2,w+kj-2] * mask[k][y,x]
// Block = (n, y, 64-wide x strip). Mask tile (25x64) + per-channel x halo
// (4 channels x 5 rows x 36 cols) staged in LDS.
// ---------------------------------------------------------------------------
__global__ __launch_bounds__(256)
void assemble_kernel(const float* __restrict__ x,
                     const float* __restrict__ maskup,
                     float* __restrict__ out)
{
    __shared__ float smask[KK][64];
    __shared__ float sx[4][5][37];

    const int xs = blockIdx.x & 1;            // x strip: 0 or 1
    const int y  = (blockIdx.x >> 1) & 127;
    const int n  = blockIdx.x >> 8;
    const int x0 = xs << 6;                   // 0 or 64
    const int w0 = x0 >> 1;                   // 0 or 32
    const int h  = y >> 1;
    const int t  = threadIdx.x;

    // load mask tile: 25 x 64 = 1600 values
    for (int e = t; e < KK * 64; e += 256) {
        const int k = e >> 6, xi = e & 63;
        smask[k][xi] = maskup[(((size_t)n * KK + k) << 14) + (y << 7) + x0 + xi];
    }

    const int ci = t >> 6;                    // channel within 4-group
    const int xi = t & 63;                    // output x within strip
    const int wl = xi >> 1;                   // local input column base

    for (int cb = 0; cb < CIN; cb += 4) {
        __syncthreads();                      // protect sx from prior readers
        // load 4 channels' halo: 4 x 5 x 36 = 720 values
        for (int e = t; e < 720; e += 256) {
            const int cl  = e / 180;
            const int rem = e - cl * 180;
            const int row = rem / 36;
            const int col = rem - row * 36;
            const int gr  = h + row - 2;
            const int gc  = w0 + col - 2;
            float v = 0.f;
            if (gr >= 0 && gr < HH && gc >= 0 && gc < WW)
                v = x[(((size_t)n * CIN + cb + cl) << 12) + gr * WW + gc];
            sx[cl][row][col] = v;
        }
        __syncthreads();

        float s = 0.f;
#pragma unroll
        for (int ki = 0; ki < 5; ++ki)
#pragma unroll
            for (int kj = 0; kj < 5; ++kj)
                s += sx[ci][ki][wl + kj] * smask[ki * 5 + kj][xi];

        out[(((size_t)n * CIN + cb + ci) << 14) + (y << 7) + x0 + xi] = s;
    }
}

// ---------------------------------------------------------------------------
extern "C" void kernel_launch(void* const* d_in, const int* in_sizes, int n_in,
                              void* d_out, int out_size, void* d_ws, size_t ws_size,
                              hipStream_t stream)
{
    const float* x      = (const float*)d_in[0];
    const float* w_comp = (const float*)d_in[1];
    const float* b_comp = (const float*)d_in[2];
    const float* w_off  = (const float*)d_in[3];
    const float* b_off  = (const float*)d_in[4];
    const float* w_ker  = (const float*)d_in[5];
    const float* b_ker  = (const float*)d_in[6];
    float* out = (float*)d_out;

    float* comp   = (float*)d_ws;                       // 2*64*4096
    float* offb   = comp + (size_t)NB * CC * PIX;       // 2*8*4096
    float* kerb   = offb + (size_t)NB * 8 * PIX;        // 2*25*4096 (softmax in place)
    float* maskup = kerb + (size_t)NB * KK * PIX;       // 2*25*128*128

    // Stage 1: 1x1 conv GEMM via f32 WMMA. 512 waves, 128 blocks x 4 waves.
    comp_wmma_kernel<<<128, 128, 0, stream>>>(x, w_comp, b_comp, comp);

    // Stage 2: 3x3 convs for off + ker.
    {
        const int total = NB * NOC * PIX;
        offker_kernel<<<(total + 255) / 256, 256, 0, stream>>>(
            comp, w_off, b_off, w_ker, b_ker, offb, kerb);
    }

    // Stage 3: softmax over 25 mask channels.
    softmax_kernel<<<(NB * PIX + 255) / 256, 256, 0, stream>>>(kerb);

    // Stage 4: grid-sample masks to output resolution.
    maskup_kernel<<<(NB * HOUT * WOUT + 255) / 256, 256, 0, stream>>>(
        offb, kerb, maskup);

    // Stage 5: final weighted 5x5 gather, LDS-tiled.
    assemble_kernel<<<NB * 128 * 2, 256, 0, stream>>>(x, maskup, out);
}